// TemporalGNN_81080392614195
// MI455X (gfx1250) — compile-verified
//
#include <hip/hip_runtime.h>
#include <hip/hip_bf16.h>

// ---------- CDNA5 WMMA types ----------
typedef __attribute__((ext_vector_type(2))) float v2f;
typedef __attribute__((ext_vector_type(8))) float v8f;

// ============================================================
// fp32 WMMA GEMM, N fixed to 64: C[M,64] = A[M,K] @ B[K,64]
//   M % 16 == 0, K % 4 == 0. One wave owns a full 16x64 strip:
//   per k-step -> 1 A load (b64) + 8 B elems + 4 x V_WMMA_F32_16X16X4_F32.
//   Exact grids so EXEC is all-ones around every WMMA (ISA 7.12 req).
// ============================================================
__global__ void gemm_wmma_f32_n64(const float* __restrict__ A,
                                  const float* __restrict__ B,
                                  float* __restrict__ C,
                                  int M, int K) {
  const int lane = threadIdx.x & 31;
  const int wave = threadIdx.x >> 5;
  const int mtile = blockIdx.x * (blockDim.x >> 5) + wave;
  if ((mtile << 4) >= M) return;  // wave-uniform

  // A operand layout (ISA 7.12.2, 32-bit A 16x4):
  //   lanes 0-15 : M = lane,    holds K+0 (v0), K+1 (v1)
  //   lanes 16-31: M = lane-16, holds K+2 (v0), K+3 (v1)
  const int m  = lane & 15;
  const int kk = (lane >> 4) << 1;
  const int n  = lane & 15;  // column within a 16-wide B/C tile

  const float* __restrict__ Arow = A + (size_t)((mtile << 4) + m) * K + kk;
  const float* __restrict__ Bn   = B + n;

  v8f acc0 = {}, acc1 = {}, acc2 = {}, acc3 = {};
  for (int k0 = 0; k0 < K; k0 += 4) {
    v2f a;
    a.x = Arow[k0];
    a.y = Arow[k0 + 1];

    // B operand (4x16 per tile): lane holds B[k0+kk][col], B[k0+kk+1][col]
    const float* __restrict__ br0 = Bn + (size_t)(k0 + kk) * 64;  // row k0+kk
    const float* __restrict__ br1 = br0 + 64;                     // row k0+kk+1
    v2f b0, b1, b2, b3;
    b0.x = br0[0];  b0.y = br1[0];
    b1.x = br0[16]; b1.y = br1[16];
    b2.x = br0[32]; b2.y = br1[32];
    b3.x = br0[48]; b3.y = br1[48];

    // 8 args: (neg_a, A, neg_b, B, c_mod, C, reuse_a, reuse_b)
    acc0 = __builtin_amdgcn_wmma_f32_16x16x4_f32(false, a, false, b0, (short)0, acc0, false, false);
    acc1 = __builtin_amdgcn_wmma_f32_16x16x4_f32(false, a, false, b1, (short)0, acc1, false, false);
    acc2 = __builtin_amdgcn_wmma_f32_16x16x4_f32(false, a, false, b2, (short)0, acc2, false, false);
    acc3 = __builtin_amdgcn_wmma_f32_16x16x4_f32(false, a, false, b3, (short)0, acc3, false, false);
  }

  // C/D layout: VGPR r -> row = r + 8*(lane>=16), col = (lane&15) + 16*tile
  const int row0 = (mtile << 4) + ((lane >> 4) << 3);
  float* __restrict__ Cout = C + (size_t)row0 * 64 + n;
#pragma unroll
  for (int r = 0; r < 8; ++r) {
    float* __restrict__ crow = Cout + (size_t)r * 64;
    crow[0]  = acc0[r];
    crow[16] = acc1[r];
    crow[32] = acc2[r];
    crow[48] = acc3[r];
  }
}

// ============================================================
// Degree / normalization helpers
// ============================================================
__global__ void deg_init(float* __restrict__ deg, int N) {
  int i = blockIdx.x * blockDim.x + threadIdx.x;
  if (i < N) deg[i] = 1.0f;  // self-loop contributes 1 to every node
}

__global__ void deg_count(const int* __restrict__ dst, float* __restrict__ deg, int E) {
  int e = blockIdx.x * blockDim.x + threadIdx.x;
  if (e < E) unsafeAtomicAdd(&deg[dst[e]], 1.0f);
}

__global__ void deg_to_dinv(float* __restrict__ deg, int N) {
  int i = blockIdx.x * blockDim.x + threadIdx.x;
  if (i < N) deg[i] = rsqrtf(deg[i]);  // deg >= 1 always (self-loops)
}

// ============================================================
// Edge scatter: agg[dst] += h[src] * dinv[src]*dinv[dst]
// thread t -> edge t>>6, feature t&63 (H == 64); h is L2-resident (25.6 MB)
// ============================================================
__global__ void scatter_edges(const int* __restrict__ src, const int* __restrict__ dst,
                              const float* __restrict__ h, const float* __restrict__ dinv,
                              float* __restrict__ agg, int E) {
  int t = blockIdx.x * blockDim.x + threadIdx.x;
  int e = t >> 6;
  if (e >= E) return;
  int f = t & 63;
  int s = src[e], d = dst[e];
  float norm = dinv[s] * dinv[d];
  float v = h[(size_t)s * 64 + f] * norm;
  unsafeAtomicAdd(&agg[(size_t)d * 64 + f], v);
}

// ============================================================
// Finalize layer: agg = relu(agg + h*dinv^2 (self-loop) + bias), in place
// ============================================================
__global__ void finalize_layer(float* __restrict__ agg, const float* __restrict__ h,
                               const float* __restrict__ dinv, const float* __restrict__ bias,
                               int total /* = N*64 */) {
  int t = blockIdx.x * blockDim.x + threadIdx.x;
  if (t >= total) return;
  int node = t >> 6, f = t & 63;
  float di = dinv[node];
  float v = agg[t] + h[t] * di * di + bias[f];
  agg[t] = fmaxf(v, 0.0f);
}

// ============================================================
// Global mean pool (sums + counts via atomics)
// ============================================================
__global__ void pool_sum(const float* __restrict__ h, const int* __restrict__ batch,
                         float* __restrict__ pooled, int total /* = N*64 */) {
  int t = blockIdx.x * blockDim.x + threadIdx.x;
  if (t >= total) return;
  int node = t >> 6, f = t & 63;
  unsafeAtomicAdd(&pooled[batch[node] * 64 + f], h[t]);
}

__global__ void pool_cnt(const int* __restrict__ batch, float* __restrict__ cnt, int N) {
  int i = blockIdx.x * blockDim.x + threadIdx.x;
  if (i < N) unsafeAtomicAdd(&cnt[batch[i]], 1.0f);
}

// ============================================================
// Head: out[g,o] = relu( (pooled[g,:]/max(cnt,1)) @ Wfc[:,o] + bfc[o] )
// G=64, H=64, OUT=32 -> 2048 threads (too small to matter; scalar FMAs)
// ============================================================
__global__ void head_fc(const float* __restrict__ pooled, const float* __restrict__ cnt,
                        const float* __restrict__ Wfc, const float* __restrict__ bfc,
                        float* __restrict__ out) {
  int t = blockIdx.x * blockDim.x + threadIdx.x;
  if (t >= 64 * 32) return;
  int g = t >> 5, o = t & 31;
  float inv = 1.0f / fmaxf(cnt[g], 1.0f);
  float s = 0.0f;
#pragma unroll 8
  for (int f = 0; f < 64; ++f)
    s += pooled[g * 64 + f] * Wfc[f * 32 + o];
  out[t] = fmaxf(s * inv + bfc[o], 0.0f);
}

// ============================================================
// Host-side orchestration (graph-capture safe: only async ops on stream)
// ============================================================
extern "C" void kernel_launch(void* const* d_in, const int* in_sizes, int n_in,
                              void* d_out, int out_size, void* d_ws, size_t ws_size,
                              hipStream_t stream) {
  (void)n_in; (void)out_size; (void)ws_size;
  constexpr int IN = 128, H = 64, G = 64;

  const float* x    = (const float*)d_in[0];
  const int*   ei   = (const int*)d_in[1];     // edge_index [2, E] flat
  const int*   batch= (const int*)d_in[2];
  const float* W1   = (const float*)d_in[3];
  const float* b1   = (const float*)d_in[4];
  const float* W2   = (const float*)d_in[5];
  const float* b2   = (const float*)d_in[6];
  const float* Wfc  = (const float*)d_in[7];
  const float* bfc  = (const float*)d_in[8];
  float* out = (float*)d_out;

  const int N = in_sizes[0] / IN;   // 100000
  const int E = in_sizes[1] / 2;    // 1200000
  const int* src = ei;
  const int* dst = ei + E;

  // Workspace carve-up (floats): ~52 MB total
  float* ws     = (float*)d_ws;
  float* dinv   = ws;                       // N
  float* hbuf   = dinv + N;                 // N*H  (GEMM output, both layers)
  float* aggbuf = hbuf + (size_t)N * H;     // N*H  (scatter target / layer input)
  float* pooled = aggbuf + (size_t)N * H;   // G*H
  float* cnt    = pooled + G * H;           // G

  const int T256 = 256;
  auto blocks = [](long long n, int b) { return (unsigned)((n + b - 1) / b); };

  // ---- degrees & normalization ----
  deg_init<<<blocks(N, T256), T256, 0, stream>>>(dinv, N);
  deg_count<<<blocks(E, T256), T256, 0, stream>>>(dst, dinv, E);
  deg_to_dinv<<<blocks(N, T256), T256, 0, stream>>>(dinv, N);

  // ---- Layer 1: h = x @ W1 (WMMA), scatter, finalize ----
  {
    const int mtiles = N / 16;               // 6250 waves, 2 waves/block
    gemm_wmma_f32_n64<<<blocks(mtiles, 2), 64, 0, stream>>>(x, W1, hbuf, N, IN);
  }
  hipMemsetAsync(aggbuf, 0, (size_t)N * H * sizeof(float), stream);
  scatter_edges<<<blocks((long long)E * H, T256), T256, 0, stream>>>(
      src, dst, hbuf, dinv, aggbuf, E);
  finalize_layer<<<blocks((long long)N * H, T256), T256, 0, stream>>>(
      aggbuf, hbuf, dinv, b1, N * H);

  // ---- Layer 2: h = a1 @ W2 (WMMA), scatter, finalize ----
  {
    const int mtiles = N / 16;
    gemm_wmma_f32_n64<<<blocks(mtiles, 2), 64, 0, stream>>>(aggbuf, W2, hbuf, N, H);
  }
  hipMemsetAsync(aggbuf, 0, (size_t)N * H * sizeof(float), stream);
  scatter_edges<<<blocks((long long)E * H, T256), T256, 0, stream>>>(
      src, dst, hbuf, dinv, aggbuf, E);
  finalize_layer<<<blocks((long long)N * H, T256), T256, 0, stream>>>(
      aggbuf, hbuf, dinv, b2, N * H);

  // ---- Global mean pool + FC head ----
  hipMemsetAsync(pooled, 0, (size_t)(G * H + G) * sizeof(float), stream);
  pool_sum<<<blocks((long long)N * H, T256), T256, 0, stream>>>(aggbuf, batch, pooled, N * H);
  pool_cnt<<<blocks(N, T256), T256, 0, stream>>>(batch, cnt, N);
  head_fc<<<blocks(G * 32, T256), T256, 0, stream>>>(pooled, cnt, Wfc, bfc, out);
}